// ScoreKernel_52664888983982
// MI455X (gfx1250) — compile-verified
//
#include <hip/hip_runtime.h>
#include <hip/hip_bf16.h>

typedef __attribute__((ext_vector_type(2))) float v2f;
typedef __attribute__((ext_vector_type(4))) float f4;
typedef __attribute__((ext_vector_type(8))) float v8f;

// One wave32 computes one 16x16 (n,r) tile:
//   G = X_tile * Z_tile^T  via 4x chained V_WMMA_F32_16X16X4_F32 (K=16)
//   d2 = |x|^2 + |z|^2 - 2G ;  k = exp(-0.5*d2/s^2)
//   out0[n,r,j] = -k*(x_j - z_j)/s^2
//   out1[n,r,j] =  k*((x_j - z_j)^2/s^4 - 1/s^2)
__global__ __launch_bounds__(256) void rbf_grad_hess_kernel(
    const float* __restrict__ X,     // [N,16]
    const float* __restrict__ Z,     // [R,16]
    const float* __restrict__ sigma, // [1]
    float* __restrict__ out,         // [2,N,R,16]
    int N, int R, int nTiles)
{
    const int lane = threadIdx.x & 31;
    const int wave = threadIdx.x >> 5;
    const int tile = blockIdx.x * 8 + wave;
    if (tile >= nTiles) return;              // uniform per wave: EXEC stays all-1s

    const int rTiles = R >> 4;
    const int tn = tile / rTiles;
    const int tr = tile - tn * rTiles;

    const int half = lane >> 4;              // 0: lanes 0-15, 1: lanes 16-31
    const int l15  = lane & 15;

    // ---- Gram tile via WMMA f32 16x16x4, chained over K=16 ----
    // A fragment: M = l15, K = kk*4 + half*2 + {0,1}
    // B fragment: N = l15, K = kk*4 + half*2 + {0,1}  (B[k][n] = Z[n][k])
    const float* __restrict__ xa = X + (size_t)(tn * 16 + l15) * 16;
    const float* __restrict__ zb = Z + (size_t)(tr * 16 + l15) * 16;

    v8f c = {};
#pragma unroll
    for (int kk = 0; kk < 4; ++kk) {
        const int kb = kk * 4 + half * 2;
        v2f a, b;
        a.x = xa[kb];  a.y = xa[kb + 1];
        b.x = zb[kb];  b.y = zb[kb + 1];
        // (neg_a, A, neg_b, B, c_mod, C, reuse_a, reuse_b)
        c = __builtin_amdgcn_wmma_f32_16x16x4_f32(false, a, false, b,
                                                  (short)0, c, false, false);
    }

    const float s      = sigma[0];
    const float inv_s2 = 1.0f / (s * s);

    // z row for this lane's column (r = tr*16 + l15, constant across the 8 C rows)
    float zrow[16];
    float zr2 = 0.0f;
#pragma unroll
    for (int j = 0; j < 16; j += 4) {
        f4 t = *(const f4*)(zb + j);
        zrow[j] = t.x; zrow[j+1] = t.y; zrow[j+2] = t.z; zrow[j+3] = t.w;
    }
#pragma unroll
    for (int j = 0; j < 16; ++j) zr2 += zrow[j] * zrow[j];

    const int    r     = tr * 16 + l15;
    const size_t NR16  = (size_t)N * (size_t)R * 16;   // offset of out[1]

#pragma unroll
    for (int v = 0; v < 8; ++v) {
        // C/D layout: VGPR v holds M = v (lanes 0-15) or M = v+8 (lanes 16-31)
        const int n = tn * 16 + v + half * 8;
        const float* __restrict__ xr = X + (size_t)n * 16;

        float xrow[16];
        float xn2 = 0.0f;
#pragma unroll
        for (int j = 0; j < 16; j += 4) {
            f4 t = *(const f4*)(xr + j);
            xrow[j] = t.x; xrow[j+1] = t.y; xrow[j+2] = t.z; xrow[j+3] = t.w;
        }
#pragma unroll
        for (int j = 0; j < 16; ++j) xn2 += xrow[j] * xrow[j];

        const float g  = c[v];
        const float d2 = xn2 + zr2 - 2.0f * g;
        const float k  = __expf(-0.5f * d2 * inv_s2);
        const float km = -k * inv_s2;                  // multiplies diff for dkdx
        const float kh =  k * inv_s2 * inv_s2;         // multiplies diff^2 for hess
        const float kc = -k * inv_s2;                  // constant hess term

        float* __restrict__ o1 = out + ((size_t)n * R + r) * 16;
        float* __restrict__ o2 = o1 + NR16;

#pragma unroll
        for (int j = 0; j < 16; j += 4) {
            f4 g1, g2;
            float d0 = xrow[j]   - zrow[j];
            float d1 = xrow[j+1] - zrow[j+1];
            float d2e = xrow[j+2] - zrow[j+2];
            float d3 = xrow[j+3] - zrow[j+3];
            g1.x = km * d0;  g1.y = km * d1;  g1.z = km * d2e;  g1.w = km * d3;
            g2.x = kh * d0 * d0  + kc;
            g2.y = kh * d1 * d1  + kc;
            g2.z = kh * d2e * d2e + kc;
            g2.w = kh * d3 * d3  + kc;
            *(f4*)(o1 + j) = g1;   // global_store_b128
            *(f4*)(o2 + j) = g2;   // global_store_b128
        }
    }
}

extern "C" void kernel_launch(void* const* d_in, const int* in_sizes, int n_in,
                              void* d_out, int out_size, void* d_ws, size_t ws_size,
                              hipStream_t stream) {
    const float* X     = (const float*)d_in[0];
    const float* Z     = (const float*)d_in[1];
    const float* sigma = (const float*)d_in[2];
    float* out = (float*)d_out;

    const int d = 16;
    const int N = in_sizes[0] / d;   // 2048
    const int R = in_sizes[1] / d;   // 1024
    const int nTiles = (N >> 4) * (R >> 4);
    const int blocks = (nTiles + 7) / 8;     // 8 waves (tiles) per 256-thread block

    rbf_grad_hess_kernel<<<blocks, 256, 0, stream>>>(X, Z, sigma, out, N, R, nTiles);
}